// LNCCLoss_32186484916609
// MI455X (gfx1250) — compile-verified
//
#include <hip/hip_runtime.h>

#define DIMX   160
#define HT     16            // output tile height
#define WT     16            // output tile width
#define DC     32            // D-slices per block
#define PAD    4
#define IH     (HT + 2*PAD)  // 24 input rows (K dim of band matmul)
#define IW     (WT + 2*PAD)  // 24 input cols
#define NFIELD 5
#define RING   9

typedef __attribute__((ext_vector_type(2))) float v2f;
typedef __attribute__((ext_vector_type(8))) float v8f;

__device__ __forceinline__ int clampi(int v, int lo, int hi) {
    return v < lo ? lo : (v > hi ? hi : v);
}

__global__ void lncc_init(float* out) { out[0] = 0.0f; }

__global__ __launch_bounds__(256)
void lncc_fused(const float* __restrict__ x, const float* __restrict__ y,
                float* __restrict__ out) {
    __shared__ float xs[IH][IW];
    __shared__ float ys[IH][IW];
    __shared__ float wsum[NFIELD][IH][WT];          // W-summed products
    __shared__ float ring[RING][NFIELD][HT][WT];    // HW-summed slice ring
    __shared__ float red[8];

    const int tid  = threadIdx.x;
    const int lane = tid & 31;
    const int wave = tid >> 5;
    const int wave_u = __builtin_amdgcn_readfirstlane(wave); // force SGPR branch

    const int ht = blockIdx.x / (DIMX / WT);
    const int wt = blockIdx.x % (DIMX / WT);
    const int h0 = ht * HT;
    const int w0 = wt * WT;
    const int d0 = blockIdx.y * DC;
    const int bb = blockIdx.z;
    const size_t plane = (size_t)DIMX * DIMX;
    const size_t base  = (size_t)bb * DIMX * plane;

    float partial = 0.0f;

    for (int it = 0; it < DC + 2 * PAD; ++it) {
        const int dz = d0 - PAD + it;
        const int zs = clampi(dz, 0, DIMX - 1);
        const int zn = clampi(dz + 1, 0, DIMX - 1);
        const size_t sbase = base + (size_t)zs * plane;
        const size_t nbase = base + (size_t)zn * plane;

        // ---- stage raw x/y tile (edge-replicated halo) into LDS ----
        for (int idx = tid; idx < IH * IW; idx += 256) {
            const int r = idx / IW, c = idx % IW;
            const int gh = clampi(h0 - PAD + r, 0, DIMX - 1);
            const int gw = clampi(w0 - PAD + c, 0, DIMX - 1);
            const size_t o = (size_t)gh * DIMX + gw;
            xs[r][c] = x[sbase + o];
            ys[r][c] = y[sbase + o];
            __builtin_prefetch(&x[nbase + o], 0, 0);  // global_prefetch next slice
            __builtin_prefetch(&y[nbase + o], 0, 0);
        }
        __syncthreads();

        // ---- W-direction 9-tap box sums of the 5 products (VALU) ----
        for (int idx = tid; idx < IH * WT; idx += 256) {
            const int r = idx / WT, c = idx % WT;
            float sx = 0.f, sy = 0.f, sxx = 0.f, syy = 0.f, sxy = 0.f;
#pragma unroll
            for (int dw = 0; dw < 9; ++dw) {
                const float xv = xs[r][c + dw];
                const float yv = ys[r][c + dw];
                sx += xv; sy += yv;
                sxx += xv * xv; syy += yv * yv; sxy += xv * yv;
            }
            wsum[0][r][c] = sx;  wsum[1][r][c] = sy;
            wsum[2][r][c] = sxx; wsum[3][r][c] = syy;
            wsum[4][r][c] = sxy;
        }
        __syncthreads();

        // ---- H-direction 9-tap box sum = Band(16x24) x S(24x16), WMMA f32 ----
        const int slot = it % RING;
        if (wave_u < NFIELD) {
            const int f = wave_u;
            const int m = lane & 15;            // A: row M / B: col N
            const int khalf = (lane >> 4) << 1; // A: lanes 16-31 hold K+2..K+3
            const int krow  = (lane >> 4);      // B: lanes 16-31 hold odd K rows
            v8f acc = {};
#pragma unroll
            for (int kc = 0; kc < IH / 4; ++kc) {
                const int k0 = kc * 4 + khalf;
                v2f a, bm;
                a.x = (k0     >= m && k0     <= m + 8) ? 1.0f : 0.0f;
                a.y = (k0 + 1 >= m && k0 + 1 <= m + 8) ? 1.0f : 0.0f;
                bm.x = wsum[f][kc * 4 + krow][m];
                bm.y = wsum[f][kc * 4 + 2 + krow][m];
                acc = __builtin_amdgcn_wmma_f32_16x16x4_f32(
                    false, a, false, bm, (short)0, acc, false, false);
            }
#pragma unroll
            for (int j = 0; j < 8; ++j) {
                const int row = (lane < 16) ? j : (j + 8);
                ring[slot][f][row][lane & 15] = acc[j];
            }
        }
        __syncthreads();

        // ---- D-direction sum + NCC once 9 slices are resident ----
        if (it >= 2 * PAD) {
            const int row = tid >> 4;   // 256 threads == 16x16 outputs
            const int col = tid & 15;
            float s0 = 0.f, s1 = 0.f, s2 = 0.f, s3 = 0.f, s4 = 0.f;
#pragma unroll
            for (int q = 0; q < RING; ++q) {
                s0 += ring[q][0][row][col];
                s1 += ring[q][1][row][col];
                s2 += ring[q][2][row][col];
                s3 += ring[q][3][row][col];
                s4 += ring[q][4][row][col];
            }
            const float inv = 1.0f / 729.0f;
            const float xm = s0 * inv, ym = s1 * inv;
            const float x2 = s2 * inv, y2 = s3 * inv, xy = s4 * inv;
            const float cross = xy - xm * ym;
            const float vx = x2 - xm * xm;
            const float vy = y2 - ym * ym;
            partial += cross * cross / (vx * vy + 1e-5f);
        }
        __syncthreads();  // protect ring slot + xs/ys before next iteration
    }

    // ---- block reduction -> single atomic per block ----
    float v = partial;
#pragma unroll
    for (int off = 16; off > 0; off >>= 1)
        v += __shfl_down(v, off, 32);
    if (lane == 0) red[wave] = v;
    __syncthreads();
    if (wave_u == 0) {
        float t = (lane < 8) ? red[lane] : 0.0f;
#pragma unroll
        for (int off = 4; off > 0; off >>= 1)
            t += __shfl_down(t, off, 32);
        if (lane == 0)
            atomicAdd(out, -t * (1.0f / (2.0f * DIMX * DIMX * DIMX)));
    }
}

extern "C" void kernel_launch(void* const* d_in, const int* in_sizes, int n_in,
                              void* d_out, int out_size, void* d_ws, size_t ws_size,
                              hipStream_t stream) {
    const float* x = (const float*)d_in[0];
    const float* y = (const float*)d_in[1];
    float* out = (float*)d_out;

    lncc_init<<<1, 1, 0, stream>>>(out);

    dim3 grid((DIMX / HT) * (DIMX / WT), DIMX / DC, 2);  // 100 x 5 x 2 = 1000 blocks
    lncc_fused<<<grid, 256, 0, stream>>>(x, y, out);
}